// WrappedMinkowskiStridedOperation_39608188403815
// MI455X (gfx1250) — compile-verified
//
#include <hip/hip_runtime.h>
#include <hip/hip_bf16.h>

// ---------------- static problem config ----------------
#define B_    8
#define NPB_  65536
#define C_    32
#define GMAX_ 256

// flat float32 output layout (concatenated in reference return order)
#define OC_OFF    0                       // out_coords [8,256,256,3]
#define MASK_OFF  1572864                 // mask       [8,256,256]
#define FEAT_OFF  2097152                 // out_feats  [8,256,256,32]
#define ALPHA_OFF 18874368                // alpha      [1]

// workspace (int) layout: [0:16) bmin, [16:32) bmax, [32:48) start, [48:64) end
typedef __attribute__((ext_vector_type(2))) float v2f;
typedef __attribute__((ext_vector_type(8))) float v8f;

__device__ __forceinline__ int fdiv2(int a) {      // Python floor-div by 2
    return (a >= 0) ? (a >> 1) : -((-a + 1) >> 1);
}

// ---- K1: per-batch min/max of coords (one block per batch) ----
__global__ __launch_bounds__(256) void k_minmax(const int* __restrict__ coords,
                                                int* __restrict__ ws) {
    __shared__ int smn0[256], smx0[256], smn1[256], smx1[256];
    const int b = blockIdx.x;
    const int t = threadIdx.x;
    int mn0 = 0x7fffffff, mx0 = -0x7fffffff, mn1 = 0x7fffffff, mx1 = -0x7fffffff;
    const int2* cp = (const int2*)coords + (size_t)b * NPB_;
    for (int p = t; p < NPB_; p += 256) {
        __builtin_prefetch(cp + p + 4096, 0, 1);   // -> global_prefetch_b8
        int2 xy = cp[p];
        mn0 = min(mn0, xy.x); mx0 = max(mx0, xy.x);
        mn1 = min(mn1, xy.y); mx1 = max(mx1, xy.y);
    }
    smn0[t] = mn0; smx0[t] = mx0; smn1[t] = mn1; smx1[t] = mx1;
    __syncthreads();
    for (int s = 128; s > 0; s >>= 1) {
        if (t < s) {
            smn0[t] = min(smn0[t], smn0[t + s]); smx0[t] = max(smx0[t], smx0[t + s]);
            smn1[t] = min(smn1[t], smn1[t + s]); smx1[t] = max(smx1[t], smx1[t + s]);
        }
        __syncthreads();
    }
    if (t == 0) {
        ws[b * 2 + 0] = smn0[0]; ws[b * 2 + 1] = smn1[0];
        ws[16 + b * 2 + 0] = smx0[0]; ws[16 + b * 2 + 1] = smx1[0];
    }
}

// ---- K2: start/end bounds per (batch, dim) ----
__global__ void k_bounds(int* __restrict__ ws) {
    const int t = threadIdx.x;
    if (t >= 16) return;
    const int b = t >> 1, d = t & 1;
    int gm = -0x7fffffff;
    for (int bb = 0; bb < B_; ++bb) gm = max(gm, ws[16 + bb * 2 + d]);
    const int i_end = gm + 1;                       // global max + i_end_off(=1)
    const int mn = ws[b * 2 + d];
    const int mx = ws[16 + b * 2 + d];
    int s = fdiv2(mn - 1) * 2; if (s < 0) s = 0;    // max(i_start=0, ...)
    int e = (fdiv2(mx + 1) + 1) * 2; if (e > i_end) e = i_end;
    ws[32 + t] = s;
    ws[48 + t] = e;
}

// ---- K3: zero the out_feats accumulation region (float4 stores) ----
__global__ __launch_bounds__(256) void k_zero(float* __restrict__ out) {
    size_t i = (size_t)blockIdx.x * 256 + threadIdx.x;   // 4,194,304 float4s
    ((float4*)(out + FEAT_OFF))[i] = make_float4(0.f, 0.f, 0.f, 0.f);
}

// ---- K4: scatter-add. One wave32 per point; lane = channel. Only even-even
//          points land on the stride-2 output grid. 128B coalesced atomics. ----
__global__ __launch_bounds__(256) void k_scatter(const int* __restrict__ coords,
                                                 const float* __restrict__ feats,
                                                 float* __restrict__ out) {
    const unsigned gid = blockIdx.x * 256u + threadIdx.x;
    const unsigned pt = gid >> 5;          // 0 .. B*NPB-1
    const unsigned c  = gid & 31;
    const int x = coords[2 * (size_t)pt];
    const int y = coords[2 * (size_t)pt + 1];
    if ((x | y) & 1) return;               // odd coordinate -> never sampled
    const unsigned b = pt >> 16;           // NPB = 65536
    const size_t idx = (size_t)FEAT_OFF +
        ((((size_t)b * GMAX_ + (x >> 1)) * GMAX_ + (y >> 1)) * C_ + c);
    atomicAdd(out + idx, feats[(size_t)pt * C_ + c]);
}

// ---- K5: mask = vx ⊗ vy via V_WMMA_F32_16X16X4_F32 (one wave per 16x16 tile),
//          plus out_coords, masked-cell zeroing, and alpha passthrough. ----
__global__ __launch_bounds__(32) void k_mask_coords(const float* __restrict__ alpha,
                                                    const int* __restrict__ ws,
                                                    float* __restrict__ out) {
    const int lane = threadIdx.x;                  // wave32: 0..31
    const int b    = blockIdx.x >> 4 >> 4;         // blockIdx.x / 256
    const int tile = blockIdx.x & 255;
    const int i0 = (tile >> 4) << 4;               // tile row origin (i)
    const int j0 = (tile & 15) << 4;               // tile col origin (j)

    const int sx = ws[32 + b * 2 + 0], ex = ws[48 + b * 2 + 0];
    const int sy = ws[32 + b * 2 + 1], ey = ws[48 + b * 2 + 1];

    const int nn = lane & 15;
    const int gi = 2 * (i0 + nn);                  // g along dim0 for M=nn
    const int gj = 2 * (j0 + nn);                  // g along dim1 for N=nn
    const float vx = (lane < 16 && gi >= sx && gi < ex) ? 1.0f : 0.0f;
    const float vy = (gj >= sy && gj < ey) ? 1.0f : 0.0f;

    // A (16x4 f32): lanes 0-15 VGPR0 hold K=0 -> vx[M]; all other K slots = 0.
    v2f a; a.x = vx; a.y = 0.0f;
    // B (4x16 f32): every row K replicated with vy[N] (N = lane&15), so the
    // product is exactly vx[M]*vy[N] regardless of B's K-to-lane layout.
    v2f bm; bm.x = vy; bm.y = vy;
    v8f cz = {};
    // (neg_a, A, neg_b, B, c_mod, C, reuse_a, reuse_b)
    v8f d = __builtin_amdgcn_wmma_f32_16x16x4_f32(false, a, false, bm,
                                                  (short)0, cz, false, false);

    // C/D layout: reg r, lanes 0-15 -> M=r, lanes 16-31 -> M=r+8; N = lane&15.
    const int N = j0 + nn;
    const int mrow = (lane >> 4) * 8;
    #pragma unroll
    for (int r = 0; r < 8; ++r) {
        const int M = i0 + r + mrow;
        const float m = d[r];
        const size_t cell = (size_t)(b * GMAX_ + M) * GMAX_ + N;
        out[MASK_OFF + cell] = m;
        float* oc = out + OC_OFF + cell * 3;
        oc[0] = (float)b; oc[1] = (float)M; oc[2] = (float)N;
        if (m == 0.0f) {                           // zero masked-out features
            float* f = out + FEAT_OFF + cell * C_;
            #pragma unroll
            for (int ch = 0; ch < C_; ++ch) f[ch] = 0.0f;
        }
    }
    if (blockIdx.x == 0 && lane == 0) out[ALPHA_OFF] = alpha[0];
}

extern "C" void kernel_launch(void* const* d_in, const int* in_sizes, int n_in,
                              void* d_out, int out_size, void* d_ws, size_t ws_size,
                              hipStream_t stream) {
    const int*   coords = (const int*)d_in[0];    // [8,65536,2] int32
    const float* feats  = (const float*)d_in[1];  // [8,65536,32] f32
    const float* alpha  = (const float*)d_in[2];  // [1] f32
    float* out = (float*)d_out;
    int*   ws  = (int*)d_ws;

    k_minmax<<<B_, 256, 0, stream>>>(coords, ws);
    k_bounds<<<1, 32, 0, stream>>>(ws);
    k_zero<<<16384, 256, 0, stream>>>(out);                 // 16.7M floats
    k_scatter<<<(B_ * NPB_ * 32) / 256, 256, 0, stream>>>(coords, feats, out);
    k_mask_coords<<<B_ * 256, 32, 0, stream>>>(alpha, ws, out);
}